// SpatialTransformer_43568148251073
// MI455X (gfx1250) — compile-verified
//
#include <hip/hip_runtime.h>

// 3D trilinear warp (spatial transformer), align_corners=True, border padding.
// src:  (B=2, C=2, D=160, H=160, W=160) f32
// flow: (B=2, 3,   D,     H,     W)     f32  (z,y,x displacement in voxels)
// out:  (B=2, C=2, D,     H,     W)     f32
//
// Memory-bound gather kernel. src (65.5MB) fits in the 192MB L2 and is reused
// ~8x by the gathers -> cache it normally. flow/out are streaming -> use
// non-temporal hints so they don't evict src from L2.

#define DD 160
#define HH 160
#define WW 160
#define NSP (DD * HH * WW)   // 4,096,000 spatial voxels
#define CC 2
#define BB 2

__global__ __launch_bounds__(WW) void warp3d_trilinear_kernel(
    const float* __restrict__ src,
    const float* __restrict__ flow,
    float* __restrict__ out)
{
    // Block = one (b, d, h) row of W voxels. All blockIdx decode is SALU.
    unsigned bz = blockIdx.x;          // 0 .. B*D*H-1
    int h = (int)(bz % HH);
    unsigned t = bz / HH;
    int d = (int)(t % DD);
    int b = (int)(t / DD);
    int w = (int)threadIdx.x;

    int s = (d * HH + h) * WW + w;     // spatial flat index

    // ---- streaming flow reads (non-temporal: protect L2 residency of src) ----
    const float* fb = flow + (size_t)b * 3 * NSP + s;
    float fz = __builtin_nontemporal_load(fb);
    float fy = __builtin_nontemporal_load(fb + NSP);
    float fx = __builtin_nontemporal_load(fb + 2 * NSP);

    // ---- coordinates: clamp(grid + flow, 0, S-1), floor, frac ----
    float cz = fminf(fmaxf((float)d + fz, 0.0f), (float)(DD - 1));
    float cy = fminf(fmaxf((float)h + fy, 0.0f), (float)(HH - 1));
    float cx = fminf(fmaxf((float)w + fx, 0.0f), (float)(WW - 1));

    float z0f = floorf(cz), y0f = floorf(cy), x0f = floorf(cx);
    float tz = cz - z0f, ty = cy - y0f, tx = cx - x0f;
    int z0 = (int)z0f, y0 = (int)y0f, x0 = (int)x0f;

    // c1 = min(c0+1, S-1) expressed as branch-free index deltas
    int dz = (z0 < DD - 1) ? (HH * WW) : 0;
    int dy = (y0 < HH - 1) ? WW : 0;
    int dx = (x0 < WW - 1) ? 1 : 0;

    int i000 = (z0 * HH + y0) * WW + x0;
    int i001 = i000 + dx;
    int i010 = i000 + dy;
    int i011 = i010 + dx;
    int i100 = i000 + dz;
    int i101 = i100 + dx;
    int i110 = i100 + dy;
    int i111 = i110 + dx;

    // ---- 8 corner weights (shared across channels) ----
    float gz = 1.0f - tz, gy = 1.0f - ty, gx = 1.0f - tx;
    float w00 = gz * gy, w01 = gz * ty, w10 = tz * gy, w11 = tz * ty;
    float w000 = w00 * gx, w001 = w00 * tx;
    float w010 = w01 * gx, w011 = w01 * tx;
    float w100 = w10 * gx, w101 = w10 * tx;
    float w110 = w11 * gx, w111 = w11 * tx;

    const float* pb = src + (size_t)b * CC * NSP;
    float* ob = out + (size_t)b * CC * NSP + s;

#pragma unroll
    for (int c = 0; c < CC; ++c) {
        const float* p = pb + (size_t)c * NSP;
        // Gathers hit L2 (src fits in 192MB); default RT caching.
        float acc;
        acc  = p[i000] * w000;
        acc  = fmaf(p[i001], w001, acc);
        acc  = fmaf(p[i010], w010, acc);
        acc  = fmaf(p[i011], w011, acc);
        acc  = fmaf(p[i100], w100, acc);
        acc  = fmaf(p[i101], w101, acc);
        acc  = fmaf(p[i110], w110, acc);
        acc  = fmaf(p[i111], w111, acc);
        // Streaming write-once output: non-temporal store.
        __builtin_nontemporal_store(acc, ob + (size_t)c * NSP);
    }
}

extern "C" void kernel_launch(void* const* d_in, const int* in_sizes, int n_in,
                              void* d_out, int out_size, void* d_ws, size_t ws_size,
                              hipStream_t stream) {
    const float* src  = (const float*)d_in[0];   // (2,2,160,160,160)
    const float* flow = (const float*)d_in[1];   // (2,3,160,160,160)
    float* out = (float*)d_out;                  // (2,2,160,160,160)

    dim3 grid(BB * DD * HH);   // 51,200 blocks, one per (b,d,h) row
    dim3 block(WW);            // 160 threads = 5 wave32 waves
    warp3d_trilinear_kernel<<<grid, block, 0, stream>>>(src, flow, out);
}